// NCELoss_3126736191777
// MI455X (gfx1250) — compile-verified
//
#include <hip/hip_runtime.h>
#include <math.h>

typedef __attribute__((ext_vector_type(2))) float v2f;
typedef __attribute__((ext_vector_type(4))) float v4f;
typedef __attribute__((ext_vector_type(8))) float v8f;
typedef __attribute__((ext_vector_type(4))) int   v4i;

#define N 8192
#define D 512
#define HALFN 4096
#define INVT 12.5f          // 1 / 0.08
#define BPAD 516            // LDS row stride in floats: 516 % 64 == 4 -> conflict-free b64 frags
#define CSPLIT 8            // column splits (blockIdx.y) -> 512 WGs, ~2 WGs/WGP for latency hiding
#define NTILES (N / 16 / CSPLIT)

#if defined(__has_builtin)
#if __has_builtin(__builtin_amdgcn_global_load_async_to_lds_b128)
#define HAVE_ASYNC_LDS 1
#endif
#endif

__device__ __forceinline__ void wait_async0() {
#if defined(__has_builtin)
#if __has_builtin(__builtin_amdgcn_s_wait_asynccnt)
    __builtin_amdgcn_s_wait_asynccnt(0);
    return;
#endif
#endif
    asm volatile("s_wait_asynccnt 0" ::: "memory");
}

// ---------------- K1: row L2-normalize: z = x / max(||x||, 1e-8) ----------------
__global__ __launch_bounds__(256) void normalize_kernel(const float* __restrict__ x,
                                                        float* __restrict__ z) {
    const int lane = threadIdx.x & 31;
    const int row  = blockIdx.x * 8 + (threadIdx.x >> 5);
    const float* xr = x + row * D;
    float ss = 0.0f;
    #pragma unroll
    for (int k = lane; k < D; k += 32) { float v = xr[k]; ss += v * v; }
    #pragma unroll
    for (int off = 1; off < 32; off <<= 1) ss += __shfl_xor(ss, off, 32);
    const float inv = 1.0f / fmaxf(sqrtf(ss), 1e-8f);
    #pragma unroll
    for (int k = lane; k < D; k += 32) z[row * D + k] = xr[k] * inv;
}

// ---------------- K2: pos[i] = (z[i] . z[(i+N/2)%N]) * (1/T) ----------------
__global__ __launch_bounds__(256) void pos_kernel(const float* __restrict__ z,
                                                  float* __restrict__ pos) {
    const int lane = threadIdx.x & 31;
    const int row  = blockIdx.x * 8 + (threadIdx.x >> 5);
    const int prow = (row + HALFN) & (N - 1);
    const float* a = z + row * D;
    const float* b = z + prow * D;
    float s = 0.0f;
    #pragma unroll
    for (int k = lane; k < D; k += 32) s += a[k] * b[k];
    #pragma unroll
    for (int off = 1; off < 32; off <<= 1) s += __shfl_xor(s, off, 32);
    if (lane == 0) pos[row] = s * INVT;
}

// ---------------- K3: fused sim-GEMM (fp32 WMMA) + streaming stats ----------------
// grid (64, CSPLIT) x 256 threads. Wave w of block bx owns rows bx*128 + w*16 .. +15.
// A panel lives in registers (128 x v2f). B tiles double-buffered in LDS, streamed in
// with GLOBAL_LOAD_ASYNC_TO_LDS_B128 (ASYNCcnt). Per-wave DS latency in the WMMA loop
// is hidden by running multiple waves per SIMD (CSPLIT=8 -> ~2 WGs/WGP).
__global__ __launch_bounds__(256, 1) void sim_stats_kernel(const float* __restrict__ z,
                                                           const float* __restrict__ pos,
                                                           float* __restrict__ se_part,
                                                           float* __restrict__ cnt_part) {
    __shared__ float ldsB[2][16 * BPAD];   // 2 x 33 KB column tiles

    const int lane = threadIdx.x & 31;
    const int wave = threadIdx.x >> 5;
    const int half = lane >> 4;            // K-pair select for A/B fragments
    const int ln   = lane & 15;
    const int rowBase = blockIdx.x * 128 + wave * 16;
    const int arow    = rowBase + ln;

    // per-thread slice of the cooperative tile fetch (8 x 16B per thread)
    int bcolv[8], bkv[8];
    #pragma unroll
    for (int j = 0; j < 8; ++j) {
        const int linear = (j * 256 + threadIdx.x) * 4;   // float index 0..8191
        bcolv[j] = linear >> 9;                           // /512
        bkv[j]   = linear & (D - 1);
    }

    // A fragment: lane ln holds row (rowBase+ln); half selects K pair (k,k+1) vs (k+2,k+3)
    v2f a[128];
    #pragma unroll
    for (int g = 0; g < 128; ++g)
        a[g] = *(const v2f*)(z + arow * D + 4 * g + 2 * half);

    float posv[8];
    #pragma unroll
    for (int m = 0; m < 8; ++m)
        posv[m] = pos[rowBase + 8 * half + m];

    float se[8], cnt[8];
    #pragma unroll
    for (int m = 0; m < 8; ++m) { se[m] = 0.0f; cnt[m] = 0.0f; }

    const int ctBase = blockIdx.y * NTILES;

    // ---- tile fetch: async global->LDS (ASYNCcnt), no VGPR staging ----
    auto issue_tile = [&](int colTile, float* buf) {
        #pragma unroll
        for (int j = 0; j < 8; ++j) {
            const float* gsrc = z + (colTile * 16 + bcolv[j]) * D + bkv[j];
            float* ldst = buf + bcolv[j] * BPAD + bkv[j];
#if defined(HAVE_ASYNC_LDS)
            __builtin_amdgcn_global_load_async_to_lds_b128(
                (__attribute__((address_space(1))) v4i*)gsrc,
                (__attribute__((address_space(3))) v4i*)ldst, 0, 0);
#else
            *(v4f*)ldst = *(const v4f*)gsrc;
#endif
        }
    };

    issue_tile(ctBase, &ldsB[0][0]);                     // prologue: stage tile 0

    for (int ct = 0; ct < NTILES; ++ct) {
        const int colTile = ctBase + ct;
        const int p = ct & 1;

#if defined(HAVE_ASYNC_LDS)
        wait_async0();                                   // my loads for tile ct done
#endif
        __syncthreads();                                 // everyone's loads done, prev reads done

        if (ct + 1 < NTILES)
            issue_tile(colTile + 1, &ldsB[p ^ 1][0]);    // overlap next tile with compute

        // ---- 128 x v_wmma_f32_16x16x4_f32 fed by ds_load_2addr_b64 ----
        const float* bbase = &ldsB[p][0] + ln * BPAD + 2 * half;
        v8f c = {0.f, 0.f, 0.f, 0.f, 0.f, 0.f, 0.f, 0.f};
        #pragma unroll
        for (int g = 0; g < 128; ++g) {
            v2f b = *(const v2f*)(bbase + 4 * g);
            c = __builtin_amdgcn_wmma_f32_16x16x4_f32(
                    false, a[g], false, b, (short)0, c, false, false);
        }
        // Scheduler steering (0x008 = WMMA group, 0x100 = DS read group): keep the
        // DS-read / WMMA cadence tight so other waves can fill the wait slots.
        __builtin_amdgcn_sched_group_barrier(0x100, 3, 0);
        #pragma unroll
        for (int g2 = 0; g2 < 64; ++g2) {
            __builtin_amdgcn_sched_group_barrier(0x008, 2, 0); // 2 WMMA
            __builtin_amdgcn_sched_group_barrier(0x100, 1, 0); // 1 DS read (2addr pair)
        }

        // ---- streaming statistics: C vgpr m <-> row rowBase + 8*half + m, lane ln <-> col ----
        const int colg = colTile * 16 + ln;
        #pragma unroll
        for (int m = 0; m < 8; ++m) {
            const int row = rowBase + 8 * half + m;
            const float val = c[m] * INVT;
            const bool diag = (colg == row);
            const bool isp  = (colg == ((row + HALFN) & (N - 1)));
            if (!diag) se[m] += __expf(val);             // diag masked to -inf -> exp 0
            cnt[m] += (!diag && !isp && (val > posv[m])) ? 1.0f : 0.0f;
        }
    }

    // reduce across the 16 lanes of each half (columns of this fragment)
    #pragma unroll
    for (int m = 0; m < 8; ++m) {
        float s = se[m], q = cnt[m];
        #pragma unroll
        for (int off = 1; off < 16; off <<= 1) {
            s += __shfl_xor(s, off, 32);
            q += __shfl_xor(q, off, 32);
        }
        se[m] = s; cnt[m] = q;
    }
    if (ln == 0) {
        #pragma unroll
        for (int m = 0; m < 8; ++m) {
            const int row = rowBase + 8 * half + m;
            se_part [blockIdx.y * N + row] = se[m];
            cnt_part[blockIdx.y * N + row] = cnt[m];
        }
    }
}

// ---------------- K4: per-row combine (deterministic order) + block nll partials ----------------
__global__ __launch_bounds__(256) void combine_kernel(const float* __restrict__ se_part,
                                                      const float* __restrict__ cnt_part,
                                                      const float* __restrict__ pos,
                                                      float* __restrict__ ranking_out,
                                                      float* __restrict__ nll_part) {
    const int row = blockIdx.x * 256 + threadIdx.x;
    float se = 0.0f, cnt = 0.0f;
    #pragma unroll
    for (int p = 0; p < CSPLIT; ++p) {
        se  += se_part [p * N + row];
        cnt += cnt_part[p * N + row];
    }
    ranking_out[row] = cnt;                       // integer-valued, exact in fp32
    const float t = -pos[row] + logf(se);         // lse (diag already excluded)

    __shared__ float red[256];
    red[threadIdx.x] = t;
    __syncthreads();
    #pragma unroll
    for (int s = 128; s > 0; s >>= 1) {
        if (threadIdx.x < s) red[threadIdx.x] += red[threadIdx.x + s];
        __syncthreads();
    }
    if (threadIdx.x == 0) nll_part[blockIdx.x] = red[0];
}

// ---------------- K5: final nll reduction (32 partials -> 1) ----------------
__global__ void nll_kernel(const float* __restrict__ nll_part, float* __restrict__ out) {
    float s = nll_part[threadIdx.x];              // exactly 32 partials, one wave
    #pragma unroll
    for (int off = 1; off < 32; off <<= 1) s += __shfl_xor(s, off, 32);
    if (threadIdx.x == 0) *out = s * (1.0f / (float)N);
}

extern "C" void kernel_launch(void* const* d_in, const int* in_sizes, int n_in,
                              void* d_out, int out_size, void* d_ws, size_t ws_size,
                              hipStream_t stream) {
    (void)in_sizes; (void)n_in; (void)out_size; (void)ws_size;
    const float* x = (const float*)d_in[0];
    float* out = (float*)d_out;                   // [0..N): ranks (float), [N]: nll
    float* ws  = (float*)d_ws;

    float* z        = ws;                           // N*D
    float* pos      = z + (size_t)N * D;            // N
    float* se_part  = pos + N;                      // CSPLIT*N
    float* cnt_part = se_part + (size_t)CSPLIT * N; // CSPLIT*N
    float* nll_part = cnt_part + (size_t)CSPLIT * N;// 32

    normalize_kernel<<<N / 8, 256, 0, stream>>>(x, z);
    pos_kernel<<<N / 8, 256, 0, stream>>>(z, pos);
    dim3 grid(N / 128, CSPLIT);
    sim_stats_kernel<<<grid, 256, 0, stream>>>(z, pos, se_part, cnt_part);
    combine_kernel<<<N / 256, 256, 0, stream>>>(se_part, cnt_part, pos, out, nll_part);
    nll_kernel<<<1, 32, 0, stream>>>(nll_part, out + N);
}